// Patchify3D_37546604101805
// MI455X (gfx1250) — compile-verified
//
#include <hip/hip_runtime.h>
#include <hip/hip_bf16.h>

// Patchify3D for MI455X (gfx1250, wave32):
//   Kernel 1: farthest-point sampling, 1 workgroup per batch, points in VGPRs.
//   Kernel 2: kNN via V_WMMA_F32_16X16X4_F32 score tiles + warp-bitonic top-32.
// Cross-lane ops use DPP16/DPP8 (VALU) for xor 1/2/4 and ds_swizzle for
// xor 8/16, replacing the ds_bpermute+s_wait_dscnt chains from round 1.

#define NPTS   16384
#define NBATCH 16
#define NCTR   512
#define KNN    32

typedef __attribute__((ext_vector_type(2))) float v2f;
typedef __attribute__((ext_vector_type(8))) float v8f;
typedef unsigned long long u64;

// ---------------------------------------------------------------------------
// lane-xor exchange primitives (wave32)
// ---------------------------------------------------------------------------
template <int MASK>
__device__ __forceinline__ unsigned int lanexor32(unsigned int v) {
#if __has_builtin(__builtin_amdgcn_update_dpp)
  if constexpr (MASK == 1)  // quad_perm [1,0,3,2]
    return (unsigned int)__builtin_amdgcn_update_dpp((int)v, (int)v, 0xB1, 0xF,
                                                     0xF, true);
  else if constexpr (MASK == 2)  // quad_perm [2,3,0,1]
    return (unsigned int)__builtin_amdgcn_update_dpp((int)v, (int)v, 0x4E, 0xF,
                                                     0xF, true);
  else
#endif
#if __has_builtin(__builtin_amdgcn_mov_dpp8)
  if constexpr (MASK == 4)  // dpp8 sel: lane i -> i^4
    return (unsigned int)__builtin_amdgcn_mov_dpp8((int)v, 0x688FAC);
  else
#endif
#if __has_builtin(__builtin_amdgcn_ds_swizzle)
  if constexpr (MASK == 8 || MASK == 16)  // group-of-32 xor swizzle, imm mask
    return (unsigned int)__builtin_amdgcn_ds_swizzle((int)v,
                                                     0x1F | (MASK << 10));
  else
#endif
    return (unsigned int)__shfl_xor((int)v, MASK, 32);
}

template <int MASK>
__device__ __forceinline__ u64 lanexor64(u64 v) {
  unsigned int lo = lanexor32<MASK>((unsigned int)v);
  unsigned int hi = lanexor32<MASK>((unsigned int)(v >> 32));
  return ((u64)hi << 32) | lo;
}

__device__ __forceinline__ u64 bcast_lane0_u64(u64 v) {
#if __has_builtin(__builtin_amdgcn_readlane)
  unsigned int lo = (unsigned int)__builtin_amdgcn_readlane((int)(unsigned int)v, 0);
  unsigned int hi = (unsigned int)__builtin_amdgcn_readlane((int)(v >> 32), 0);
  return ((u64)hi << 32) | lo;
#else
  return __shfl(v, 0, 32);
#endif
}

// compare-exchange with lane lane^MASK; keep min in "lower" lane iff dirUp
template <int MASK>
__device__ __forceinline__ u64 cswap(u64 v, bool dirUp, int lane) {
  u64 o = lanexor64<MASK>(v);
  bool lower = (lane & MASK) == 0;
  u64 mn = (v < o) ? v : o;
  u64 mx = (v < o) ? o : v;
  return (lower == dirUp) ? mn : mx;
}

// full bitonic sort, ascending across lanes 0..31
__device__ __forceinline__ u64 bitonic_sort32_asc(u64 v, int lane) {
  v = cswap<1>(v, (lane & 2) == 0, lane);    // k=2
  v = cswap<2>(v, (lane & 4) == 0, lane);    // k=4
  v = cswap<1>(v, (lane & 4) == 0, lane);
  v = cswap<4>(v, (lane & 8) == 0, lane);    // k=8
  v = cswap<2>(v, (lane & 8) == 0, lane);
  v = cswap<1>(v, (lane & 8) == 0, lane);
  v = cswap<8>(v, (lane & 16) == 0, lane);   // k=16
  v = cswap<4>(v, (lane & 16) == 0, lane);
  v = cswap<2>(v, (lane & 16) == 0, lane);
  v = cswap<1>(v, (lane & 16) == 0, lane);
  v = cswap<16>(v, true, lane);              // k=32
  v = cswap<8>(v, true, lane);
  v = cswap<4>(v, true, lane);
  v = cswap<2>(v, true, lane);
  v = cswap<1>(v, true, lane);
  return v;
}

// merge a bitonic sequence into sorted order (dirUp: ascending)
__device__ __forceinline__ u64 bitonic_merge32(u64 v, int lane, bool dirUp) {
  v = cswap<16>(v, dirUp, lane);
  v = cswap<8>(v, dirUp, lane);
  v = cswap<4>(v, dirUp, lane);
  v = cswap<2>(v, dirUp, lane);
  v = cswap<1>(v, dirUp, lane);
  return v;
}

// butterfly max-reduce (result in all lanes)
__device__ __forceinline__ u64 wave_max_u64(u64 v) {
  u64 o;
  o = lanexor64<16>(v); v = (o > v) ? o : v;
  o = lanexor64<8>(v);  v = (o > v) ? o : v;
  o = lanexor64<4>(v);  v = (o > v) ? o : v;
  o = lanexor64<2>(v);  v = (o > v) ? o : v;
  o = lanexor64<1>(v);  v = (o > v) ? o : v;
  return v;
}

// order-preserving float->u32 map (handles negative scores)
__device__ __forceinline__ unsigned int fmap(float f) {
  unsigned int b = __float_as_uint(f);
  return (b & 0x80000000u) ? ~b : (b | 0x80000000u);
}

// ---------------------------------------------------------------------------
// Kernel 1: FPS. grid = NBATCH blocks, 1024 threads. Each thread owns 16
// points in registers (strided). Packed argmax key = dist_bits<<32 | ~idx
// (max dist, lowest index on ties, like jnp.argmax).
// ---------------------------------------------------------------------------
__global__ __launch_bounds__(1024) void fps_kernel(
    const float* __restrict__ x, float* __restrict__ centers) {
  const int b = blockIdx.x;
  const float* pts = x + (size_t)b * NPTS * 3;
  const int tid = threadIdx.x;
  const int lane = tid & 31, wave = tid >> 5;

  float px[16], py[16], pz[16], dmin[16];
#pragma unroll
  for (int i = 0; i < 16; ++i) {
    int p = tid + (i << 10);  // strided: coalesced b96 loads
    px[i] = pts[p * 3 + 0];
    py[i] = pts[p * 3 + 1];
    pz[i] = pts[p * 3 + 2];
    dmin[i] = 1e10f;
  }

  __shared__ float s_last[3];
  __shared__ u64 s_wavemax[32];
  __shared__ u64 s_winner;

  if (tid == 0) {  // first selected index is 0
    s_last[0] = pts[0];
    s_last[1] = pts[1];
    s_last[2] = pts[2];
  }
  __syncthreads();

  for (int it = 0; it < NCTR; ++it) {
    float lx = s_last[0], ly = s_last[1], lz = s_last[2];
    if (tid == 0) {  // record center = coords of current 'last'
      centers[((size_t)b * NCTR + it) * 3 + 0] = lx;
      centers[((size_t)b * NCTR + it) * 3 + 1] = ly;
      centers[((size_t)b * NCTR + it) * 3 + 2] = lz;
    }
    u64 best = 0ull;
#pragma unroll
    for (int i = 0; i < 16; ++i) {
      float dx = px[i] - lx, dy = py[i] - ly, dz = pz[i] - lz;
      float d = fmaf(dx, dx, fmaf(dy, dy, dz * dz));
      dmin[i] = fminf(dmin[i], d);
      unsigned int idx = (unsigned int)(tid + (i << 10));
      u64 key = ((u64)__float_as_uint(dmin[i]) << 32) | (0xFFFFFFFFu - idx);
      best = (key > best) ? key : best;
    }
    best = wave_max_u64(best);
    if (lane == 0) s_wavemax[wave] = best;
    __syncthreads();
    if (wave == 0) {
      u64 v = wave_max_u64(s_wavemax[lane]);
      if (lane == 0) s_winner = v;
    }
    __syncthreads();
    if (tid == 0) {  // reload winner coords from global (L2 hit)
      unsigned int widx = 0xFFFFFFFFu - (unsigned int)s_winner;
      s_last[0] = pts[widx * 3 + 0];
      s_last[1] = pts[widx * 3 + 1];
      s_last[2] = pts[widx * 3 + 2];
    }
    __syncthreads();
  }
}

// ---------------------------------------------------------------------------
// Kernel 2: kNN + gather. grid = NBATCH*32 blocks (16 centers each),
// 256 threads = 8 waves. Per chunk of 128 points each wave runs one
// V_WMMA_F32_16X16X4_F32 producing a 16(center)x16(point) score tile
// (score = ||p||^2 - 2 c.p, rank-equivalent to d^2 per row), spills it to
// LDS, then each wave maintains running top-32 for 2 center rows via
// warp bitonic merge with ballot early-skip.
//
// Operand layout assumption for f32 16x16x4 (per ISA 7.12.2 striping):
//   A (16x4, 2 VGPRs): lanes0-15 -> K={0,1}, lanes16-31 -> K={2,3}; M = lane&15
//   B (4x16, 2 VGPRs): lanes0-15 -> K={0,1}, lanes16-31 -> K={2,3}; N = lane&15
//   D (16x16, 8 VGPRs): D[v], lanes0-15 -> M=v, lanes16-31 -> M=v+8; N = lane&15
// ---------------------------------------------------------------------------
#define SC_STRIDE 132  // padded to dodge LDS bank conflicts on tile writes

__global__ __launch_bounds__(256) void knn_kernel(
    const float* __restrict__ x, const float* __restrict__ centers,
    float* __restrict__ out) {
  const int b = blockIdx.x >> 5;   // batch
  const int mt = blockIdx.x & 31;  // center tile (16 centers)
  const float* pts = x + (size_t)b * NPTS * 3;
  const int tid = threadIdx.x, lane = tid & 31, wave = tid >> 5;
  const int half = lane >> 4, l16 = lane & 15;

  __shared__ float s_scores[16][SC_STRIDE];
  __shared__ float s_ctr[48];

  if (tid < 48) s_ctr[tid] = centers[((size_t)b * NCTR + mt * 16) * 3 + tid];
  __syncthreads();

  // A matrix: row m = (-2cx, -2cy, -2cz, 1)
  v2f A;
  if (half == 0) {
    A.x = -2.f * s_ctr[l16 * 3 + 0];  // K=0
    A.y = -2.f * s_ctr[l16 * 3 + 1];  // K=1
  } else {
    A.x = -2.f * s_ctr[l16 * 3 + 2];  // K=2
    A.y = 1.0f;                       // K=3
  }

  // running top-32 keys for rows 2*wave (curA) and 2*wave+1 (curB),
  // kept sorted DESCENDING across lanes (lane 0 = 32nd best = threshold)
  u64 curA = ~0ull, curB = ~0ull;

  for (int chunk = 0; chunk < NPTS / 128; ++chunk) {
    const int base = chunk * 128;
    const int n = base + wave * 16 + l16;
    const float* p = pts + (size_t)n * 3;
    float vx = p[0], vy = p[1], vz = p[2];
    if (chunk + 1 < NPTS / 128)
      __builtin_prefetch(pts + (size_t)(n + 128) * 3, 0, 3);  // near-cache
    float nrm = fmaf(vx, vx, fmaf(vy, vy, vz * vz));

    // B matrix: col n = (x, y, z, ||p||^2)
    v2f Bm;
    if (half == 0) {
      Bm.x = vx;   // K=0
      Bm.y = vy;   // K=1
    } else {
      Bm.x = vz;   // K=2
      Bm.y = nrm;  // K=3
    }

    v8f C = {};
    C = __builtin_amdgcn_wmma_f32_16x16x4_f32(
        /*neg_a=*/false, A, /*neg_b=*/false, Bm,
        /*c_mod=*/(short)0, C, /*reuse_a=*/false, /*reuse_b=*/false);

#pragma unroll
    for (int v = 0; v < 8; ++v)
      s_scores[v + half * 8][wave * 16 + l16] = C[v];
    __syncthreads();

    // selection: wave handles rows 2*wave, 2*wave+1; 4 groups of 32 candidates
#pragma unroll
    for (int r = 0; r < 2; ++r) {
      int row = wave * 2 + r;
      u64 cur = r ? curB : curA;
#pragma unroll
      for (int g = 0; g < 4; ++g) {
        float d = s_scores[row][g * 32 + lane];
        unsigned int idx = (unsigned int)(base + g * 32 + lane);
        u64 key = ((u64)fmap(d) << 32) | idx;
        u64 thresh = bcast_lane0_u64(cur);  // current 32nd best (readlane)
        if (__ballot(key < thresh)) {
          key = bitonic_sort32_asc(key, lane);        // new ascending
          u64 m = (key < cur) ? key : cur;            // 32 smallest, bitonic
          cur = bitonic_merge32(m, lane, /*dirUp=*/false);  // restore desc
        }
      }
      if (r) curB = cur; else curA = cur;
    }
    __syncthreads();
  }

  // output: sort ascending (desc sequence is bitonic -> one asc merge),
  // lane l = l-th nearest neighbor; gather coords.
#pragma unroll
  for (int r = 0; r < 2; ++r) {
    int row = wave * 2 + r;
    u64 cur = r ? curB : curA;
    cur = bitonic_merge32(cur, lane, /*dirUp=*/true);
    unsigned int idx = (unsigned int)cur;
    const float* p = pts + (size_t)idx * 3;
    int mg = mt * 16 + row;
    size_t o = (((size_t)b * NCTR + mg) * KNN + lane) * 3;
    out[o + 0] = p[0];
    out[o + 1] = p[1];
    out[o + 2] = p[2];
  }
}

// ---------------------------------------------------------------------------
extern "C" void kernel_launch(void* const* d_in, const int* in_sizes, int n_in,
                              void* d_out, int out_size, void* d_ws,
                              size_t ws_size, hipStream_t stream) {
  const float* x = (const float*)d_in[0];        // [16,16384,3] f32
  float* centers = (float*)d_ws;                 // [16,512,3] f32 scratch
  float* out = (float*)d_out;                    // [16,512,32,3] f32

  fps_kernel<<<NBATCH, 1024, 0, stream>>>(x, centers);
  knn_kernel<<<NBATCH * 32, 256, 0, stream>>>(x, centers, out);
}